// RelationAttention_2044404433792
// MI455X (gfx1250) — compile-verified
//
#include <hip/hip_runtime.h>
#include <hip/hip_bf16.h>

#define N_NODES 50000
#define IN_DIM 128
#define OUT_DIM 128
#define H 8
#define DK 16
#define SLOPE 0.2f

typedef float v2f __attribute__((ext_vector_type(2)));
typedef float v8f __attribute__((ext_vector_type(8)));

// ---------------------------------------------------------------------------
// Fold relation_att into Wk:  Wk_rel[(h,e),c] = sum_d rel[h,d,e] * Wk[(h,d),c]
// and bk_rel[(h,e)] = sum_d rel[h,d,e] * bk[(h,d)].
// ---------------------------------------------------------------------------
__global__ __launch_bounds__(256)
void fuse_wk_kernel(const float* __restrict__ Wk, const float* __restrict__ bk,
                    const float* __restrict__ rel,
                    float* __restrict__ Wk_rel, float* __restrict__ bk_rel) {
    int idx = blockIdx.x * blockDim.x + threadIdx.x;
    if (idx >= OUT_DIM * IN_DIM) return;
    int row = idx >> 7;          // output row = h*16 + e
    int col = idx & 127;
    int h = row >> 4;
    int e = row & 15;
    float s = 0.f;
    #pragma unroll
    for (int d = 0; d < DK; ++d)
        s += rel[(h * DK + d) * DK + e] * Wk[(h * DK + d) * IN_DIM + col];
    Wk_rel[row * IN_DIM + col] = s;
    if (col == 0) {
        float sb = 0.f;
        #pragma unroll
        for (int d = 0; d < DK; ++d)
            sb += rel[(h * DK + d) * DK + e] * bk[h * DK + d];
        bk_rel[row] = sb;
    }
}

// ---------------------------------------------------------------------------
// C(M x 128) = X(M x 128) @ W(128 x 128)^T + bias, fp32 WMMA 16x16x4.
// One wave per 16x16 output tile. M is a multiple of 16 -> EXEC all ones.
//
// ISA lane layouts (CDNA5 ISA):
//   A 16x4:  lane L -> M = L%16; VGPR0 = K(2*(L/16)), VGPR1 = K(2*(L/16)+1)
//   B 4x16:  lane L -> N = L%16; VGPR0 = K(2*(L/16)), VGPR1 = K(2*(L/16)+1)
//   C/D:     c[r] = D[r + 8*(L/16)][L%16]
// ---------------------------------------------------------------------------
__global__ __launch_bounds__(32)
void gemm_xwT_128(const float* __restrict__ X, const float* __restrict__ W,
                  const float* __restrict__ bias, float* __restrict__ C) {
    const int lane = threadIdx.x & 31;
    const int mt   = blockIdx.x;          // M/16 tiles
    const int nt   = blockIdx.y;          // 128/16 = 8 tiles
    const int half = lane >> 4;           // 0 or 1
    const int l16  = lane & 15;
    const float* xrow = X + (size_t)(mt * 16 + l16) * IN_DIM;  // A row (M)
    const float* wrow = W + (size_t)(nt * 16 + l16) * IN_DIM;  // B col source (W row n)

    v8f c = {};
    #pragma unroll
    for (int kb = 0; kb < IN_DIM; kb += 4) {
        const int k0 = kb + 2 * half;
        v2f a, b;
        a.x = xrow[k0]; a.y = xrow[k0 + 1];
        b.x = wrow[k0]; b.y = wrow[k0 + 1];
        // D = A(16x4) * B(4x16) + C   -> v_wmma_f32_16x16x4_f32
        c = __builtin_amdgcn_wmma_f32_16x16x4_f32(
                /*neg_a=*/false, a, /*neg_b=*/false, b,
                /*c_mod=*/(short)0, c, /*reuse_a=*/false, /*reuse_b=*/false);
    }

    const float bn = bias[nt * 16 + l16];
    #pragma unroll
    for (int r = 0; r < 8; ++r) {
        const int row = mt * 16 + r + 8 * half;
        C[(size_t)row * OUT_DIM + nt * 16 + l16] = c[r] + bn;
    }
}

// ---------------------------------------------------------------------------
// Per-edge, per-head logits:  dot(q[dst,h,:], k'[src,h,:]) / 4, leaky-relu.
// ---------------------------------------------------------------------------
__global__ __launch_bounds__(256)
void edge_logits_kernel(const float* __restrict__ q, const float* __restrict__ kk,
                        const int* __restrict__ src_idx, const int* __restrict__ dst_idx,
                        float* __restrict__ logits, int E) {
    int e = blockIdx.x * blockDim.x + threadIdx.x;
    if (e >= E) return;
    const float4* qr = (const float4*)(q  + (size_t)dst_idx[e] * OUT_DIM);
    const float4* kr = (const float4*)(kk + (size_t)src_idx[e] * OUT_DIM);
    #pragma unroll
    for (int h = 0; h < H; ++h) {
        float s = 0.f;
        #pragma unroll
        for (int i = 0; i < 4; ++i) {
            float4 a = qr[h * 4 + i];
            float4 b = kr[h * 4 + i];
            s += a.x * b.x + a.y * b.y + a.z * b.z + a.w * b.w;
        }
        s *= 0.25f;                                    // 1/sqrt(16)
        logits[(size_t)e * H + h] = (s >= 0.f) ? s : SLOPE * s;
    }
}

// ---------------------------------------------------------------------------
// Segment softmax + weighted aggregation. dst_idx is sorted: each block binary
// searches its edge range (deterministic, no atomics). 8 waves = 8 heads.
// ---------------------------------------------------------------------------
__device__ __forceinline__ int lower_bound_i32(const int* __restrict__ a, int n, int key) {
    int lo = 0, hi = n;
    while (lo < hi) {
        int mid = (lo + hi) >> 1;
        if (a[mid] < key) lo = mid + 1; else hi = mid;
    }
    return lo;
}

__global__ __launch_bounds__(256)
void dst_aggregate_kernel(const float* __restrict__ logits, const float* __restrict__ v,
                          const int* __restrict__ src_idx, const int* __restrict__ dst_idx,
                          float* __restrict__ hout, int E) {
    const int d    = blockIdx.x;
    const int head = threadIdx.x >> 5;
    const int lane = threadIdx.x & 31;

    const int lo = lower_bound_i32(dst_idx, E, d);
    const int hi = lower_bound_i32(dst_idx, E, d + 1);

    if (lo >= hi) {   // no incoming edges -> zero row
        if (lane < DK) hout[(size_t)d * OUT_DIM + head * DK + lane] = 0.f;
        return;
    }

    // pass 1: segment max per head
    float m = -1e30f;
    for (int e = lo + lane; e < hi; e += 32)
        m = fmaxf(m, logits[(size_t)e * H + head]);
    #pragma unroll
    for (int off = 16; off > 0; off >>= 1)
        m = fmaxf(m, __shfl_xor(m, off, 32));

    // pass 2: sum of exp and exp-weighted V accumulation
    float s = 0.f;
    float acc[DK];
    #pragma unroll
    for (int i = 0; i < DK; ++i) acc[i] = 0.f;

    for (int e = lo + lane; e < hi; e += 32) {
        float ex = __expf(logits[(size_t)e * H + head] - m);
        s += ex;
        const float4* vr = (const float4*)(v + (size_t)src_idx[e] * OUT_DIM + head * DK);
        #pragma unroll
        for (int i = 0; i < 4; ++i) {
            float4 vv = vr[i];
            acc[4*i+0] += ex * vv.x;
            acc[4*i+1] += ex * vv.y;
            acc[4*i+2] += ex * vv.z;
            acc[4*i+3] += ex * vv.w;
        }
    }

    #pragma unroll
    for (int off = 16; off > 0; off >>= 1) {
        s += __shfl_xor(s, off, 32);
        #pragma unroll
        for (int i = 0; i < DK; ++i)
            acc[i] += __shfl_xor(acc[i], off, 32);
    }

    if (lane < DK)
        hout[(size_t)d * OUT_DIM + head * DK + lane] = acc[lane] / s;
}

// ---------------------------------------------------------------------------
// Launch
// ---------------------------------------------------------------------------
extern "C" void kernel_launch(void* const* d_in, const int* in_sizes, int n_in,
                              void* d_out, int out_size, void* d_ws, size_t ws_size,
                              hipStream_t stream) {
    const float* feat_src = (const float*)d_in[0];
    const float* feat_dst = (const float*)d_in[1];
    const int*   src_idx  = (const int*)d_in[2];
    const int*   dst_idx  = (const int*)d_in[3];
    const float* Wq       = (const float*)d_in[4];
    const float* bq       = (const float*)d_in[5];
    const float* Wk       = (const float*)d_in[6];
    const float* bk       = (const float*)d_in[7];
    const float* Wv       = (const float*)d_in[8];
    const float* bv       = (const float*)d_in[9];
    const float* rel      = (const float*)d_in[10];
    const float* Wo       = (const float*)d_in[11];
    const float* bo       = (const float*)d_in[12];
    float* out = (float*)d_out;

    const int E = in_sizes[2];

    // workspace layout (floats)
    float* ws = (float*)d_ws;
    const size_t NF = (size_t)N_NODES * OUT_DIM;       // 6.4M floats
    float* qbuf   = ws;                                // 25.6 MB
    float* kbuf   = ws + NF;                           // 25.6 MB
    float* vbuf   = ws + 2 * NF;                       // 25.6 MB
    float* hbuf   = ws + 3 * NF;                       // 25.6 MB
    float* logits = ws + 4 * NF;                       // E*8*4 = 25.6 MB
    float* Wk_rel = ws + 4 * NF + (size_t)E * H;       // 64 KB
    float* bk_rel = Wk_rel + OUT_DIM * IN_DIM;         // 512 B

    // 1. fold relation_att into Wk
    fuse_wk_kernel<<<(OUT_DIM * IN_DIM + 255) / 256, 256, 0, stream>>>(
        Wk, bk, rel, Wk_rel, bk_rel);

    // 2-4. WMMA projections: q, k' (relation-fused), v
    dim3 ggrid(N_NODES / 16, OUT_DIM / 16);
    gemm_xwT_128<<<ggrid, 32, 0, stream>>>(feat_dst, Wq,     bq,     qbuf);
    gemm_xwT_128<<<ggrid, 32, 0, stream>>>(feat_src, Wk_rel, bk_rel, kbuf);
    gemm_xwT_128<<<ggrid, 32, 0, stream>>>(feat_src, Wv,     bv,     vbuf);

    // 5. per-edge attention logits (leaky relu applied)
    edge_logits_kernel<<<(E + 255) / 256, 256, 0, stream>>>(
        qbuf, kbuf, src_idx, dst_idx, logits, E);

    // 6. segment softmax + weighted aggregation (one block per dst node)
    dst_aggregate_kernel<<<N_NODES, 256, 0, stream>>>(
        logits, vbuf, src_idx, dst_idx, hbuf, E);

    // 7. output projection into d_out
    gemm_xwT_128<<<ggrid, 32, 0, stream>>>(hbuf, Wo, bo, out);
}